// part_14199161881179
// MI455X (gfx1250) — compile-verified
//
#include <hip/hip_runtime.h>

typedef __attribute__((ext_vector_type(2))) float v2f;
typedef __attribute__((ext_vector_type(8))) float v8f;
typedef int v4i_vs __attribute__((vector_size(4 * sizeof(int))));  // matches builtin param

#define NROWS 4096
#define DCOLS 512
#define OUTSTRIDE 1024
#define LDS_STRIDE 72   // 64 cols + pad (keeps 16B alignment: 72*4=288=18*16)

// Async memory->LDS staging path (CDNA5 GLOBAL_LOAD_ASYNC_TO_LDS_B128,
// tracked by ASYNCcnt). Guarded: falls back to load+ds_store if the
// builtins are absent on this toolchain.
#if defined(__has_builtin)
#if __has_builtin(__builtin_amdgcn_global_load_async_to_lds_b128) && \
    __has_builtin(__builtin_amdgcn_s_wait_asynccnt)
#define USE_ASYNC_LDS 1
#endif
#endif

// Computes output rows [64, 4096):
//   out[i, 0:512]    = sum_{j=i-64..i} X[j, :]   (banded-ones WMMA)
//   out[i, 512:1024] = X[i, :]
// Grid: (8 col-blocks of 64, 63 row-blocks of 64). Block: 512 thr = 16 waves.
__global__ __launch_bounds__(512) void slidesum_wmma_kernel(
    const float* __restrict__ X, float* __restrict__ out)
{
    __shared__ float lds[128 * LDS_STRIDE];

    const int i0  = 64 + (int)blockIdx.y * 64;   // first output row of block
    const int c0g = (int)blockIdx.x * 64;        // first column of block
    const int tid = (int)threadIdx.x;

    // ---- stage X rows [i0-64, i0+64) x cols [c0g, c0g+64) into LDS ----
    // 128 rows x 16 float4/row = 2048 b128 transfers; 512 threads x 4 each.
    {
        #pragma unroll
        for (int k = 0; k < 4; ++k) {
            int idx = tid + k * 512;
            int row = idx >> 4;          // 0..127
            int c4  = idx & 15;          // 0..15
            const float* gsrc = X + (size_t)(i0 - 64 + row) * DCOLS + c0g + c4 * 4;
            float*       ldst = &lds[row * LDS_STRIDE + c4 * 4];
#if USE_ASYNC_LDS
            v4i_vs* gp = (v4i_vs*)const_cast<float*>(gsrc);
            v4i_vs* lp = (v4i_vs*)ldst;
            __builtin_amdgcn_global_load_async_to_lds_b128(
                (__attribute__((address_space(1))) v4i_vs*)gp,
                (__attribute__((address_space(3))) v4i_vs*)lp,
                /*imm offset=*/0, /*cpol=*/0);
#else
            *reinterpret_cast<float4*>(ldst) =
                *reinterpret_cast<const float4*>(gsrc);
#endif
        }
    }
#if USE_ASYNC_LDS
    __builtin_amdgcn_s_wait_asynccnt(0);   // this wave's async copies done
#endif
    __syncthreads();                       // make LDS visible across waves

    // ---- WMMA: each wave computes one 16x16 tile of H ----
    const int wave = tid >> 5;      // 0..15
    const int lane = tid & 31;
    const int rt   = wave >> 2;     // row tile 0..3
    const int ct   = wave & 3;      // col tile 0..3
    const int r0   = rt * 16;       // LDS base row of this tile's 80-row window
    const int c0   = ct * 16;       // LDS base col
    const int m    = lane & 15;     // M index (A) == N index (B/C/D)
    const int h    = lane >> 4;     // half-select

    v8f acc = {};
    #pragma unroll
    for (int c = 0; c < 20; ++c) {
        const int k0 = c * 4 + 2 * h;      // K covered by this lane's 2 VGPRs
        v2f a, b;
        // Band: A[m, kk] = 1 iff m <= kk <= m+64  (65-row inclusive window)
        a.x = (k0     >= m && k0     <= m + 64) ? 1.0f : 0.0f;
        a.y = (k0 + 1 >= m && k0 + 1 <= m + 64) ? 1.0f : 0.0f;
        // B[k, n] from LDS (n == m for this lane)
        b.x = lds[(r0 + k0    ) * LDS_STRIDE + c0 + m];
        b.y = lds[(r0 + k0 + 1) * LDS_STRIDE + c0 + m];
        acc = __builtin_amdgcn_wmma_f32_16x16x4_f32(
                  /*neg_a=*/false, a, /*neg_b=*/false, b,
                  /*c_mod=*/(short)0, acc, /*reuse_a=*/false, /*reuse_b=*/false);
    }

    // ---- store H half: out[i, 0:512] ----
    {
        const int it = i0 + rt * 16;
        const int dg = c0g + c0 + m;
        #pragma unroll
        for (int v = 0; v < 8; ++v) {
            int row = it + v + 8 * h;               // C layout: M=v (+8 for hi half)
            out[(size_t)row * OUTSTRIDE + dg] = acc[v];
        }
    }

    // ---- copy half: out[i, 512+d] = X[i, d], served from LDS rows 64..127 ----
    #pragma unroll
    for (int k = 0; k < 8; ++k) {
        int flat = tid + k * 512;
        int row  = flat >> 6;   // 0..63
        int col  = flat & 63;
        out[(size_t)(i0 + row) * OUTSTRIDE + 512 + c0g + col] =
            lds[(row + 64) * LDS_STRIDE + col];
    }
}

// Rows [0, 64): out[i, 0:512] = 2*X[i]; out[i, 512:1024] = X[i].
__global__ __launch_bounds__(128) void head_rows_kernel(
    const float* __restrict__ X, float* __restrict__ out)
{
    const int i  = (int)blockIdx.x;      // 0..63
    const int d4 = (int)threadIdx.x;     // 0..127 -> float4 column
    const float4* Xg = reinterpret_cast<const float4*>(X);
    float4* Og       = reinterpret_cast<float4*>(out);
    float4 v = Xg[(size_t)i * (DCOLS / 4) + d4];
    float4 v2 = make_float4(2.0f * v.x, 2.0f * v.y, 2.0f * v.z, 2.0f * v.w);
    Og[(size_t)i * (OUTSTRIDE / 4) + d4]                = v2;  // H half
    Og[(size_t)i * (OUTSTRIDE / 4) + (DCOLS / 4) + d4]  = v;   // copy half
}

extern "C" void kernel_launch(void* const* d_in, const int* in_sizes, int n_in,
                              void* d_out, int out_size, void* d_ws, size_t ws_size,
                              hipStream_t stream)
{
    (void)in_sizes; (void)n_in; (void)out_size; (void)d_ws; (void)ws_size;
    const float* X = (const float*)d_in[0];   // (4096, 512)
    // d_in[1] (W) is provably dead: softmax over a singleton axis == 1.
    float* out = (float*)d_out;               // (4096, 1024)

    dim3 grid(8, 63);                          // 8 col-blocks x 63 row-blocks
    slidesum_wmma_kernel<<<grid, 512, 0, stream>>>(X, out);
    head_rows_kernel<<<64, 128, 0, stream>>>(X, out);
}